// KMeansInference_33749853012288
// MI455X (gfx1250) — compile-verified
//
#include <hip/hip_runtime.h>
#include <hip/hip_bf16.h>
#include <math.h>

#define NPTS 262144
#define DIM  768
#define KCLU 1024
#define BM   128   // feature rows per block (8 waves x 16)
#define BN   128   // clusters per accumulation group
#define BK   32    // D-chunk staged in LDS
#define LSTR 36    // padded LDS row stride (floats): bank-conflict-free, 16B aligned

typedef __attribute__((ext_vector_type(2))) float v2f;
typedef __attribute__((ext_vector_type(8))) float v8f;

// Async global->LDS 16-byte copy (CDNA5): LDS dest addr in VGPR, 64-bit global
// addr in VGPR pair, tracked by ASYNCcnt. Bypasses the register file.
__device__ __forceinline__ void async_copy_b128(unsigned lds_off, const void* gptr) {
  asm volatile("global_load_async_to_lds_b128 %0, %1, off"
               :: "v"(lds_off), "v"((unsigned long long)(uintptr_t)gptr)
               : "memory");
}
__device__ __forceinline__ void wait_async0() {
  asm volatile("s_wait_asynccnt 0" ::: "memory");
}

// ---- c2[k] = sum_d centers[k][d]^2 ----
__global__ __launch_bounds__(256) void c2_kernel(const float* __restrict__ centers,
                                                 float* __restrict__ c2) {
  const int k = blockIdx.x;
  float s = 0.f;
  for (int d = threadIdx.x; d < DIM; d += 256) {
    float v = centers[(size_t)k * DIM + d];
    s = fmaf(v, v, s);
  }
  #pragma unroll
  for (int m = 16; m >= 1; m >>= 1) s += __shfl_xor(s, m, 32);
  __shared__ float red[8];
  if ((threadIdx.x & 31) == 0) red[threadIdx.x >> 5] = s;
  __syncthreads();
  if (threadIdx.x == 0) {
    float t = 0.f;
    #pragma unroll
    for (int i = 0; i < 8; ++i) t += red[i];
    c2[k] = t;
  }
}

// ---- fused GEMM (fp32 WMMA) + argmin ----
__global__ __launch_bounds__(256) void kmeans_argmin_kernel(
    const float* __restrict__ features, const float* __restrict__ centers,
    const float* __restrict__ c2, int* __restrict__ out) {
  __shared__ float Als[BM * LSTR];
  __shared__ float Bls[BN * LSTR];

  const int tid   = threadIdx.x;
  const int wave  = tid >> 5;
  const int lane  = tid & 31;
  const int lrow  = lane & 15;          // row/col within 16-lane half
  const int khalf = (lane >> 4) * 2;    // K offset per ISA A/B fragment layout
  const int m0    = blockIdx.x * BM;

  // coalesced tile loaders: 32 rows x 32 floats per pass, 4 passes
  const int ldr = tid >> 3;             // 0..31
  const int ldc = (tid & 7) * 4;        // 0,4,...,28

  float bestVal[8];
  int   bestIdx[8];
  #pragma unroll
  for (int v = 0; v < 8; ++v) { bestVal[v] = INFINITY; bestIdx[v] = 0; }

  for (int ct = 0; ct < KCLU / BN; ++ct) {
    v8f acc[BN / 16];
    #pragma unroll
    for (int n = 0; n < BN / 16; ++n) acc[n] = (v8f){0.f,0.f,0.f,0.f,0.f,0.f,0.f,0.f};

    for (int dk = 0; dk < DIM; dk += BK) {
      __syncthreads();  // previous iteration's LDS reads retired
      #pragma unroll
      for (int p = 0; p < 4; ++p) {
        const int r = ldr + p * 32;
        async_copy_b128((unsigned)(uintptr_t)&Als[r * LSTR + ldc],
                        &features[(size_t)(m0 + r) * DIM + dk + ldc]);
        async_copy_b128((unsigned)(uintptr_t)&Bls[r * LSTR + ldc],
                        &centers[(size_t)(ct * BN + r) * DIM + dk + ldc]);
      }
      wait_async0();    // async LDS writes landed
      __syncthreads();  // visible to all waves

      #pragma unroll
      for (int kk = 0; kk < BK / 4; ++kk) {
        const v2f a = *(const v2f*)&Als[(wave * 16 + lrow) * LSTR + kk * 4 + khalf];
        #pragma unroll
        for (int n = 0; n < BN / 16; ++n) {
          const v2f b = *(const v2f*)&Bls[(n * 16 + lrow) * LSTR + kk * 4 + khalf];
          acc[n] = __builtin_amdgcn_wmma_f32_16x16x4_f32(
              /*neg_a=*/false, a, /*neg_b=*/false, b,
              /*c_mod=*/(short)0, acc[n], /*reuse_a=*/false, /*reuse_b=*/false);
        }
      }
    }

    // epilogue for this cluster group: cand = c2[col] - 2*dot  (||x||^2 dropped)
    #pragma unroll
    for (int n = 0; n < BN / 16; ++n) {
      const int col = ct * BN + n * 16 + lrow;
      const float c2v = c2[col];
      #pragma unroll
      for (int v = 0; v < 8; ++v) {
        const float cand = fmaf(-2.0f, acc[n][v], c2v);
        if (cand < bestVal[v]) { bestVal[v] = cand; bestIdx[v] = col; }
      }
    }
  }

  // cross-lane argmin within each 16-lane half (wave32), tie -> smaller index
  #pragma unroll
  for (int v = 0; v < 8; ++v) {
    float bv = bestVal[v];
    int   bi = bestIdx[v];
    #pragma unroll
    for (int m = 1; m <= 8; m <<= 1) {
      const float ov = __shfl_xor(bv, m, 32);
      const int   oi = __shfl_xor(bi, m, 32);
      if (ov < bv || (ov == bv && oi < bi)) { bv = ov; bi = oi; }
    }
    // VGPR v: lanes 0-15 hold row v, lanes 16-31 hold row v+8 (ISA C/D layout)
    if (lrow == 0) {
      const int row = m0 + wave * 16 + v + ((lane >> 4) ? 8 : 0);
      out[row] = bi;
    }
  }
}

extern "C" void kernel_launch(void* const* d_in, const int* in_sizes, int n_in,
                              void* d_out, int out_size, void* d_ws, size_t ws_size,
                              hipStream_t stream) {
  const float* features = (const float*)d_in[0];
  const float* centers  = (const float*)d_in[1];
  float* c2  = (float*)d_ws;       // KCLU floats of scratch
  int*   out = (int*)d_out;        // argmin indices (int32)

  c2_kernel<<<KCLU, 256, 0, stream>>>(centers, c2);
  kmeans_argmin_kernel<<<NPTS / BM, 256, 0, stream>>>(features, centers, c2, out);
}